// GraphPool_7971459301496
// MI455X (gfx1250) — compile-verified
//
#include <hip/hip_runtime.h>
#include <stdint.h>

// GraphPool: out[i] = x[i] + sum_{e : edge_src[e]==i} x[edge_dst[e]]
// N_NODES=50000, N_EDGES=800000, D_FEAT=128 (fp32).
// Memory-bound random row gather -> use CDNA5 Tensor Data Mover in GATHER mode
// (8 indexed 512B rows per descriptor, async into LDS, TENSORcnt tracked),
// double-buffered, one wave32 workgroup per node.

#define GP_NODES 50000
#define GP_DFEAT 128
#define GP_ROWS  8            // rows per TDM gather op (32-bit index mode max)

typedef __attribute__((ext_vector_type(4))) unsigned int v4u;
typedef __attribute__((ext_vector_type(8))) int          v8i;
typedef __attribute__((ext_vector_type(4))) int          v4i;
typedef __attribute__((ext_vector_type(4))) float        f4;

// Issue one TDM gather: 'cnt' rows of x (row = GP_DFEAT f32) selected by
// 32-bit row indices i0..i7, landing contiguously at LDS byte offset lds_off.
__device__ __forceinline__ void tdm_gather_rows(
    const float* __restrict__ x, unsigned lds_off,
    int i0, int i1, int i2, int i3, int i4, int i5, int i6, int i7, int cnt)
{
    const uint64_t ga = (uint64_t)(uintptr_t)x;   // tensor base (gather indices select rows)

    v4u g0;
    g0.x = 1u               // count = 1 valid descriptor
         | (1u << 30)       // gather_index_size = 32-bit
         | (1u << 31);      // gather_mode = 1
    g0.y = lds_off;                                        // lds_addr (bytes)
    g0.z = (unsigned)ga;                                   // global_addr[31:0]
    g0.w = (unsigned)((ga >> 32) & 0x01FFFFFFu)            // global_addr[56:32]
         | (2u << 30);                                     // type = 2 ("image")

    v8i g1;
    g1[0] = (int)(2u << 16);                               // data_size = 4 bytes; wg_mask=0
    g1[1] = (int)(((unsigned)GP_DFEAT & 0xFFFFu) << 16);   // tensor_dim0[15:0] @ [31:16]
    g1[2] = (int)((((unsigned)GP_DFEAT >> 16) & 0xFFFFu)   // tensor_dim0[31:16]
         |        (((unsigned)GP_NODES & 0xFFFFu) << 16)); // tensor_dim1[15:0]
    g1[3] = (int)((((unsigned)GP_NODES >> 16) & 0xFFFFu)   // tensor_dim1[31:16]
         |        (((unsigned)GP_DFEAT & 0xFFFFu) << 16)); // tile_dim0 = 128
    g1[4] = cnt;                                           // tile_dim1 = #valid indices
    g1[5] = GP_DFEAT;                                      // tensor_dim0_stride[31:0] = 128
    g1[6] = 0;                                             // stride hi / dim1_stride (ignored)
    g1[7] = 0;

    v4i g2 = { i0, i1, i2, i3 };                           // row_index_0..3 (32-bit mode)
    v4i g3 = { i4, i5, i6, i7 };                           // row_index_4..7

    const v8i g4 = { 0, 0, 0, 0, 0, 0, 0, 0 };             // extra operand (clang-23 form)

    __builtin_amdgcn_tensor_load_to_lds(g0, g1, g2, g3, g4, 0);
}

__global__ __launch_bounds__(32)
void GraphPool_tdm_kernel(const float* __restrict__ x,
                          const int*   __restrict__ esrc,
                          const int*   __restrict__ edst,
                          float*       __restrict__ out,
                          int nEdges)
{
    __shared__ float sh[2][GP_ROWS * GP_DFEAT];   // 2 x 4KB double buffer

    const int node = (int)blockIdx.x;
    const int lane = (int)threadIdx.x;

    // Uniform (scalar) binary searches over sorted edge_src: [e0, e0+total)
    int lo = 0, hi = nEdges;
    while (lo < hi) { int m = (lo + hi) >> 1; if (esrc[m] <  node)     lo = m + 1; else hi = m; }
    int lo2 = lo, hi2 = nEdges;
    while (lo2 < hi2) { int m = (lo2 + hi2) >> 1; if (esrc[m] < node + 1) lo2 = m + 1; else hi2 = m; }
    const int e0 = lo;
    const int total = lo2 - lo;

    // Identity term: lane l owns features [4l, 4l+4) of this node's row.
    f4 acc = ((const f4*)(x + (size_t)node * GP_DFEAT))[lane];

    const unsigned lds0 = (unsigned)(uintptr_t)&sh[0][0];
    const unsigned lds1 = (unsigned)(uintptr_t)&sh[1][0];

    const int nChunks = (total + GP_ROWS - 1) / GP_ROWS;

    auto issue = [&](int c) {
        const int base = e0 + c * GP_ROWS;
        int cnt = total - c * GP_ROWS;
        if (cnt > GP_ROWS) cnt = GP_ROWS;
        int id[GP_ROWS];
#pragma unroll
        for (int t = 0; t < GP_ROWS; ++t) {
            const int tt = (t < cnt) ? t : (cnt - 1);   // clamp: repeats are legal
            id[t] = edst[base + tt];                    // uniform -> scalar loads
        }
        tdm_gather_rows(x, (c & 1) ? lds1 : lds0,
                        id[0], id[1], id[2], id[3], id[4], id[5], id[6], id[7], cnt);
    };

    if (nChunks > 0) issue(0);

    for (int c = 0; c < nChunks; ++c) {
        if (c + 1 < nChunks) {
            issue(c + 1);                                  // keep 2 TDM ops in flight
            __builtin_amdgcn_s_wait_tensorcnt(1);          // chunk c landed in LDS
        } else {
            __builtin_amdgcn_s_wait_tensorcnt(0);
        }
        asm volatile("" ::: "memory");                     // no LDS-read hoisting past the wait

        const float* buf = (c & 1) ? &sh[1][0] : &sh[0][0];
        int cnt = total - c * GP_ROWS;
        if (cnt > GP_ROWS) cnt = GP_ROWS;
#pragma unroll 4
        for (int r = 0; r < cnt; ++r) {
            const f4 v = ((const f4*)(buf + r * GP_DFEAT))[lane];  // ds_load_b128, conflict-free
            acc.x += v.x; acc.y += v.y; acc.z += v.z; acc.w += v.w;
        }
        asm volatile("" ::: "memory");                     // buffer reuse fence (double buffer)
    }

    ((f4*)(out + (size_t)node * GP_DFEAT))[lane] = acc;    // global_store_b128
}

extern "C" void kernel_launch(void* const* d_in, const int* in_sizes, int n_in,
                              void* d_out, int out_size, void* d_ws, size_t ws_size,
                              hipStream_t stream)
{
    const float* x    = (const float*)d_in[0];
    const int*   esrc = (const int*)  d_in[1];
    const int*   edst = (const int*)  d_in[2];
    float*       out  = (float*)      d_out;
    const int nEdges  = in_sizes[1];

    GraphPool_tdm_kernel<<<dim3(GP_NODES), dim3(32), 0, stream>>>(x, esrc, edst, out, nEdges);
}